// Nerflets_24223615549820
// MI455X (gfx1250) — compile-verified
//
#include <hip/hip_runtime.h>
#include <math.h>

typedef float v2f __attribute__((ext_vector_type(2)));
typedef float v8f __attribute__((ext_vector_type(8)));

#define NEXP  64      // experts
#define TOPK  4       // experts per point
#define WID   128     // hidden width
#define NB    1024    // batch points
#define INXYZ 63
#define INDIR 27
#define TILE  16      // rows per WMMA tile
#define STR   208     // LDS row stride (floats) for activation buffers
#define XSTR  64      // LDS row stride for xyz buffer
#define SPLIT 4       // workgroups per expert

struct MlpParams {
  const float* x;            // (B, 90)
  const float* xw[8];        // layer i: (N, in_i, 128)
  const float* xb[8];        // (N, 128)
  const float* fw; const float* fb;   // final: (N,128,128)/(N,128)
  const float* dw; const float* db;   // dir:   (N,155,64)/(N,64)
  const float* sw; const float* sb;   // sigma: (N,128,1)/(N,1)
  const float* rw; const float* rb;   // rgb:   (N,64,3)/(N,3)
  const int* lists;          // (N, B) assignment ids
  const int* counts;         // (N)
  float* out;                // (K, B, 4)
};

// ---------------- top-k (matches jax.lax.top_k tie-breaking) ----------------
__global__ void topk_kernel(const float* __restrict__ rbfs, int* __restrict__ idx) {
  int b = blockIdx.x * blockDim.x + threadIdx.x;
  if (b >= NB) return;
  float s[NEXP];
  #pragma unroll
  for (int n = 0; n < NEXP; n++) s[n] = rbfs[n * NB + b];  // rbfs is (N,B,1)
  int sel0 = -1, sel1 = -1, sel2 = -1, sel3 = -1;
  #pragma unroll
  for (int k = 0; k < TOPK; k++) {
    float best = -INFINITY; int bi = 0;
    #pragma unroll
    for (int n = 0; n < NEXP; n++) {
      if (n == sel0 || n == sel1 || n == sel2 || n == sel3) continue;
      if (s[n] > best) { best = s[n]; bi = n; }   // strict > keeps earliest index
    }
    if      (k == 0) sel0 = bi;
    else if (k == 1) sel1 = bi;
    else if (k == 2) sel2 = bi;
    else             sel3 = bi;
    idx[b * TOPK + k] = bi;
  }
}

__global__ void zero_counts_kernel(int* counts) {
  if (threadIdx.x < NEXP) counts[threadIdx.x] = 0;
}

__global__ void build_lists_kernel(const int* __restrict__ idx, int* __restrict__ counts,
                                   int* __restrict__ lists) {
  int a = blockIdx.x * blockDim.x + threadIdx.x;   // assignment id = b*K + k
  if (a >= NB * TOPK) return;
  int e = idx[a];
  int pos = atomicAdd(&counts[e], 1);
  lists[e * NB + pos] = a;
}

// ---------------- one FC layer tile: wave-level f32 WMMA GEMM ----------------
// 16 x ODIM = (16 x KDIM) * (KDIM x ODIM). Wave `wave` owns output cols [16w, 16w+16).
// All strides/dims compile-time -> B-fragment loads become base + immediate offset.
template <int KDIM, int ODIM, int SSTR, int DSTR, bool RELU>
__device__ __forceinline__ void fc_layer(int wave, int lane,
                                         const float* __restrict__ src,
                                         const float* __restrict__ w,     // expert weight base
                                         const float* __restrict__ bias,  // expert bias base
                                         float* __restrict__ dst) {
  const int n0 = wave * 16;
  if (n0 < ODIM) {                        // uniform per wave -> EXEC all ones inside
    const int half = (lane >> 4) & 1;     // range-obvious for constant folding
    const int m    = lane & 15;
    const int n    = n0 + m;
    // per-lane base: row (2*half), col n
    const float* wl = w + (size_t)(2 * half) * ODIM + n;
    constexpr int KPAD = (KDIM + 3) & ~3;
    v8f acc0 = {}, acc1 = {};             // two chains -> independent WMMA streams

    #pragma unroll
    for (int k0 = 0; k0 < KPAD; k0 += 8) {
      {
        const int kk = k0 + 2 * half;
        v2f a, bm;
        a.x = src[m * SSTR + kk];
        a.y = src[m * SSTR + kk + 1];
        if (k0 + 4 <= KDIM) {             // compile-time per unrolled iteration
          bm.x = wl[k0 * ODIM];
          bm.y = wl[(k0 + 1) * ODIM];
        } else {
          bm.x = (kk < KDIM) ? wl[k0 * ODIM] : 0.0f;
          bm.y = (kk + 1 < KDIM) ? wl[(k0 + 1) * ODIM] : 0.0f;
        }
        acc0 = __builtin_amdgcn_wmma_f32_16x16x4_f32(false, a, false, bm,
                                                     (short)0, acc0, false, false);
      }
      if (k0 + 4 < KPAD) {
        const int k1 = k0 + 4;
        const int kk = k1 + 2 * half;
        v2f a, bm;
        a.x = src[m * SSTR + kk];
        a.y = src[m * SSTR + kk + 1];
        if (k1 + 4 <= KDIM) {
          bm.x = wl[k1 * ODIM];
          bm.y = wl[(k1 + 1) * ODIM];
        } else {
          bm.x = (kk < KDIM) ? wl[k1 * ODIM] : 0.0f;
          bm.y = (kk + 1 < KDIM) ? wl[(k1 + 1) * ODIM] : 0.0f;
        }
        acc1 = __builtin_amdgcn_wmma_f32_16x16x4_f32(false, a, false, bm,
                                                     (short)0, acc1, false, false);
      }
    }

    const float bv = bias[n];
    #pragma unroll
    for (int v = 0; v < 8; v++) {         // C/D layout: VGPR v -> row v+8*half, col n0+m
      float val = acc0[v] + acc1[v] + bv;
      if (RELU) val = fmaxf(val, 0.0f);
      dst[(v + 8 * half) * DSTR + n0 + m] = val;
    }
  }
  __syncthreads();
}

// ---------------- expert MLP ----------------
__global__ void __launch_bounds__(256) moe_mlp_kernel(MlpParams p) {
  const int e    = blockIdx.x;
  const int tid  = threadIdx.x;
  const int wave = tid >> 5;
  const int lane = tid & 31;

  __shared__ float XB[TILE][XSTR];   // xyz input tile (col 63 zero-pad)
  __shared__ float BA[TILE][STR];
  __shared__ float BB[TILE][STR];
  __shared__ int   AIDS[TILE];
  __shared__ int   BROW[TILE];

  const int cnt = p.counts[e];

  // hoist per-expert weight/bias base pointers (one 64-bit offset each, once)
  const float* xw[8]; const float* xb[8];
  #pragma unroll
  for (int i = 0; i < 8; i++) {
    const int kin = (i == 0) ? INXYZ : ((i == 4) ? (INXYZ + WID) : WID);
    xw[i] = p.xw[i] + (size_t)e * kin * WID;
    xb[i] = p.xb[i] + (size_t)e * WID;
  }
  const float* fw = p.fw + (size_t)e * WID * WID;
  const float* fb = p.fb + (size_t)e * WID;
  const float* dw = p.dw + (size_t)e * (WID + INDIR) * (WID / 2);
  const float* db = p.db + (size_t)e * (WID / 2);
  const float* sw = p.sw + (size_t)e * WID;
  const float* rw = p.rw + (size_t)e * (WID / 2) * 3;

  for (int tile = blockIdx.y; tile * TILE < cnt; tile += SPLIT) {
    const int base = tile * TILE;
    if (tid < TILE) {
      int aid = (base + tid < cnt) ? p.lists[e * NB + base + tid] : -1;
      AIDS[tid] = aid;
      BROW[tid] = (aid >= 0) ? (aid / TOPK) : 0;
    }
    __syncthreads();

    // gather xyz rows (zero-pad col 63 and invalid rows)
    for (int i = tid; i < TILE * XSTR; i += 256) {
      int r = i >> 6, c = i & 63;
      float v = 0.0f;
      if (c < INXYZ && AIDS[r] >= 0) v = p.x[BROW[r] * (INXYZ + INDIR) + c];
      XB[r][c] = v;
    }
    __syncthreads();

    fc_layer<INXYZ, WID, XSTR, STR, true>(wave, lane, &XB[0][0], xw[0], xb[0], &BA[0][0]); // L0
    fc_layer<WID,   WID, STR,  STR, true>(wave, lane, &BA[0][0], xw[1], xb[1], &BB[0][0]); // L1
    fc_layer<WID,   WID, STR,  STR, true>(wave, lane, &BB[0][0], xw[2], xb[2], &BA[0][0]); // L2
    fc_layer<WID,   WID, STR,  STR, true>(wave, lane, &BA[0][0], xw[3], xb[3], &BB[0][0]); // L3 -> h in BB

    // skip concat: BA = [xyz(63) | h(128) | 0]
    for (int i = tid; i < TILE * 192; i += 256) {
      int r = i / 192, c = i % 192;
      float v;
      if (c < INXYZ)            v = XB[r][c];
      else if (c < INXYZ + WID) v = BB[r][c - INXYZ];
      else                      v = 0.0f;
      BA[r][c] = v;
    }
    __syncthreads();

    fc_layer<INXYZ + WID, WID, STR, STR, true>(wave, lane, &BA[0][0], xw[4], xb[4], &BB[0][0]); // L4
    fc_layer<WID, WID, STR, STR, true>(wave, lane, &BB[0][0], xw[5], xb[5], &BA[0][0]);         // L5
    fc_layer<WID, WID, STR, STR, true>(wave, lane, &BA[0][0], xw[6], xb[6], &BB[0][0]);         // L6
    fc_layer<WID, WID, STR, STR, true>(wave, lane, &BB[0][0], xw[7], xb[7], &BA[0][0]);         // L7 -> h in BA

    // feat = h @ final_w (no act): BA -> BB[:, 0:128]
    fc_layer<WID, WID, STR, STR, false>(wave, lane, &BA[0][0], fw, fb, &BB[0][0]);

    // sigma head (wave 0, lanes 0..15): ssoftplus(h . sw + sb)  [BA still intact]
    if (wave == 0 && lane < TILE) {
      int r = lane;
      if (AIDS[r] >= 0) {
        float s = p.sb[e];
        for (int c = 0; c < WID; c++) s += BA[r][c] * sw[c];
        float t = s - 1.0f;
        float sp = (t > 20.0f) ? t : log1pf(expf(t));
        int aid = AIDS[r], b = aid / TOPK, k = aid % TOPK;
        p.out[(size_t)k * NB * 4 + b * 4 + 3] = sp;
      }
    }

    // append view dirs: BB[:, 128:155] = dir, cols 155..159 zeroed
    for (int i = tid; i < TILE * 32; i += 256) {
      int r = i >> 5, c = i & 31;
      float v = 0.0f;
      if (c < INDIR && AIDS[r] >= 0) v = p.x[BROW[r] * (INXYZ + INDIR) + INXYZ + c];
      BB[r][WID + c] = v;
    }
    __syncthreads();

    // dir layer: (155 -> 64, relu): BB -> BA[:, 0:64]  (waves 0..3 active)
    fc_layer<WID + INDIR, WID / 2, STR, STR, true>(wave, lane, &BB[0][0], dw, db, &BA[0][0]);

    // rgb head (wave 0): sigmoid(dir_enc @ rgb_w + rb), 16 rows x 3 cols
    if (wave == 0) {
      for (int i = lane; i < TILE * 3; i += 32) {
        int r = i / 3, c = i % 3;
        if (AIDS[r] >= 0) {
          float s = p.rb[e * 3 + c];
          for (int j = 0; j < WID / 2; j++) s += BA[r][j] * rw[j * 3 + c];
          float sg = 1.0f / (1.0f + expf(-s));
          int aid = AIDS[r], b = aid / TOPK, k = aid % TOPK;
          p.out[(size_t)k * NB * 4 + b * 4 + c] = sg;
        }
      }
    }
    __syncthreads();   // protect LDS reuse by next tile
  }
}

extern "C" void kernel_launch(void* const* d_in, const int* in_sizes, int n_in,
                              void* d_out, int out_size, void* d_ws, size_t ws_size,
                              hipStream_t stream) {
  (void)in_sizes; (void)n_in; (void)out_size; (void)ws_size;
  const float* x    = (const float*)d_in[0];
  const float* rbfs = (const float*)d_in[1];

  MlpParams p;
  p.x = x;
  for (int i = 0; i < 8; i++) {
    p.xw[i] = (const float*)d_in[2 + i];
    p.xb[i] = (const float*)d_in[10 + i];
  }
  p.fw = (const float*)d_in[18]; p.fb = (const float*)d_in[19];
  p.dw = (const float*)d_in[20]; p.db = (const float*)d_in[21];
  p.sw = (const float*)d_in[22]; p.sb = (const float*)d_in[23];
  p.rw = (const float*)d_in[24]; p.rb = (const float*)d_in[25];

  int* ws_idx    = (int*)d_ws;              // B*K
  int* ws_counts = ws_idx + NB * TOPK;      // N
  int* ws_lists  = ws_counts + NEXP;        // N*B
  p.lists = ws_lists; p.counts = ws_counts;
  p.out = (float*)d_out;

  topk_kernel<<<NB / 128, 128, 0, stream>>>(rbfs, ws_idx);
  zero_counts_kernel<<<1, 64, 0, stream>>>(ws_counts);
  build_lists_kernel<<<(NB * TOPK) / 256, 256, 0, stream>>>(ws_idx, ws_counts, ws_lists);
  moe_mlp_kernel<<<dim3(NEXP, SPLIT), 256, 0, stream>>>(p);
}